// mpNet_Constrained_72438918414850
// MI455X (gfx1250) — compile-verified
//
#include <hip/hip_runtime.h>
#include <math.h>

// ---------------------------------------------------------------------------
// Fused matching-pursuit (mpNet) for MI455X / gfx1250.
//   B=1024 samples, m=16, n=64, A=1200 atoms, complex64.
// One workgroup (256 thr = 8 wave32) per sample:
//   Phase 0 (tiny kernel): bake W into WMMA-B-fragment order (L2 resident).
//   Phase 1: M_D = conj(M)^T W via V_WMMA_F32_16X16X4_F32. Two A-tiles per
//            wave iteration share A fragments -> 8 independent WMMA
//            accumulate chains; complex combine done once per tile in VALU.
//            Columns normalized in-register, stored to LDS (b128 stores).
//   Phase 2: k MP iterations entirely out of LDS (b128 loads); residual in
//            VGPRs; wave32 shuffle argmax + cross-wave LDS reduction.
// M[b] is staged global->LDS with GLOBAL_LOAD_ASYNC_TO_LDS_B128 (ASYNCcnt).
// M_D (157 MB aggregate) never touches global memory.
// ---------------------------------------------------------------------------

typedef __attribute__((ext_vector_type(2))) float v2f;
typedef __attribute__((ext_vector_type(8))) float v8f;

#define N_B     1024
#define MDIM    16
#define NANT    64
#define AATOMS  1200
#define NTILES  75        // 1200 / 16 output tiles along A
#define NPAIRS  37        // tile pairs; tile 74 is the remainder
#define KSTEPS  16        // 64 / 4 k-steps per tile
#define ASTRIDE 18        // float2 per atom column (16 data + 2 pad) = 144 B
#define THREADS 256
#define NWAVES  8
#define WFRAG   (KSTEPS * 2 * 32)   // v2f slots per tile in packed W

// shared memory layout (in floats)
#define SM_MRE   0                               // M real  [m][n] : 1024
#define SM_MIM   1024                            // M imag  [m][n] : 1024
#define SM_MD    2048                            // dictionary: A*ASTRIDE float2
#define SM_STAGE SM_MD                           // async staging (pre-phase1)
#define SM_RED   (2048 + AATOMS * ASTRIDE * 2)   // reduction scratch : 64
#define SM_TOT   (SM_RED + 64)                   // 45312 floats = 181248 B

#define WMMA4(acc, A, Bm)                                                     \
  acc = __builtin_amdgcn_wmma_f32_16x16x4_f32(false, (A), false, (Bm),        \
                                              (short)0, (acc), false, false)

// --------------------------------------------------------------------------
// Kernel 0: pack W into B-fragment order.
// Fragment (tile t, kstep s, plane p in {re,im}), lane L holds the two f32
// B elements {B(k,a), B(k+1,a)} with k = 2*(L/16), a = t*16 + L%16,
// n = 4*s + k. One float2 per (frag, lane) slot.
// --------------------------------------------------------------------------
__global__ void pack_w_kernel(const float* __restrict__ pos,
                              const float* __restrict__ doa,
                              const float* __restrict__ g,
                              const float* __restrict__ lam,
                              float2* __restrict__ wpk) {
  int idx = blockIdx.x * blockDim.x + threadIdx.x;
  if (idx >= NTILES * KSTEPS * 2 * 32) return;
  int L = idx & 31;
  int rest = idx >> 5;
  int p = rest & 1;  rest >>= 1;
  int s = rest & 15;
  int t = rest >> 4;
  int a  = t * 16 + (L & 15);
  int n0 = 4 * s + 2 * (L >> 4);
  float c  = 6.28318530717958647692f / lam[0];
  float sa = sinf(doa[a]);
  float ph0 = c * pos[n0]     * sa;
  float ph1 = c * pos[n0 + 1] * sa;
  float2 o;
  if (p == 0) { o.x = g[n0] * cosf(ph0); o.y = g[n0 + 1] * cosf(ph1); }
  else        { o.x = g[n0] * sinf(ph0); o.y = g[n0 + 1] * sinf(ph1); }
  wpk[idx] = o;
}

// --------------------------------------------------------------------------
// Combine 4 accumulate chains into a complex tile, normalize columns, store.
// C/D layout: VGPR r, lanes 0-15 hold M=r, lanes 16-31 hold M=r+8; col=a%16.
// --------------------------------------------------------------------------
__device__ __forceinline__ void finalize_tile(float2* md, int t, int lane,
                                              const v8f& rr, const v8f& ii,
                                              const v8f& ri, const v8f& ir) {
  float re[8], im[8];
  float s2 = 0.f;
#pragma unroll
  for (int r = 0; r < 8; ++r) {
    re[r] = rr[r] + ii[r];   //  Mre^T Wre + Mim^T Wim
    im[r] = ri[r] - ir[r];   //  Mre^T Wim - Mim^T Wre
    s2 += re[r] * re[r] + im[r] * im[r];
  }
  s2 += __shfl_xor(s2, 16, 32);      // partner half-lane: other 8 m-values
  float inv = 1.0f / sqrtf(s2);
  int a  = t * 16 + (lane & 15);
  int mb = 8 * (lane >> 4);
  float4* dst = (float4*)&md[a * ASTRIDE + mb];   // 16B aligned
#pragma unroll
  for (int r = 0; r < 8; r += 2) {
    float4 v;
    v.x = re[r] * inv;     v.y = im[r] * inv;
    v.z = re[r + 1] * inv; v.w = im[r + 1] * inv;
    dst[r >> 1] = v;                               // ds_store_b128
  }
}

// --------------------------------------------------------------------------
// Kernel 1: fused per-sample dictionary build + matching pursuit.
// --------------------------------------------------------------------------
__global__ __launch_bounds__(THREADS)
void mp_fused_kernel(const float2* __restrict__ Mg,   // [B, n=64, m=16] cplx
                     const float2* __restrict__ xg,   // [B, m=16] cplx
                     const v2f*    __restrict__ wpk,  // packed W fragments
                     const int*    __restrict__ kp,   // iteration count
                     float2*       __restrict__ outp) // residual | x_hat
{
  extern __shared__ float smem[];
  float*  mre = smem + SM_MRE;
  float*  mim = smem + SM_MIM;
  float2* md  = (float2*)(smem + SM_MD);
  float*  red = smem + SM_RED;

  const int b    = blockIdx.x;
  const int tid  = threadIdx.x;
  const int lane = tid & 31;
  const int wave = tid >> 5;

  // ---- async-copy M[b] (8 KB) global -> LDS staging (no VGPR roundtrip) ---
  {
    unsigned long long gaddr =
        (unsigned long long)(const void*)(Mg + (size_t)b * (NANT * MDIM)) +
        (unsigned long long)(tid * 16);
    // flat LDS pointer: low 32 bits are the wave-relative LDS byte offset
    unsigned laddr = (unsigned)(size_t)(smem + SM_STAGE) + (unsigned)(tid * 16);
    asm volatile(
        "global_load_async_to_lds_b128 %0, %1, off\n\t"
        "global_load_async_to_lds_b128 %0, %1, off offset:4096\n\t"
        "s_wait_asynccnt 0"
        :: "v"(laddr), "v"(gaddr) : "memory");
  }
  __syncthreads();

  // ---- transpose staging [n][m] cplx -> de-interleaved planes [m][n] ------
  {
    const float2* stage = (const float2*)(smem + SM_STAGE);
    float2 tmp[4];
#pragma unroll
    for (int j = 0; j < 4; ++j) tmp[j] = stage[tid + j * THREADS];
    __syncthreads();   // staging fully read before planes overwrite nothing,
                       // but md region reuse below needs all reads done
#pragma unroll
    for (int j = 0; j < 4; ++j) {
      int i = tid + j * THREADS;
      int n = i >> 4, m = i & 15;
      mre[m * NANT + n] = tmp[j].x;
      mim[m * NANT + n] = tmp[j].y;
    }
  }
  __syncthreads();

  // ---- Phase 1: M_D tiles via fp32 WMMA (2 tiles / wave-iter) -------------
  const int m16   = lane & 15;
  const int khalf = lane >> 4;
  const v8f z8 = {0.f, 0.f, 0.f, 0.f, 0.f, 0.f, 0.f, 0.f};

  for (int p = wave; p < NPAIRS; p += NWAVES) {
    const int t0 = 2 * p, t1 = 2 * p + 1;
    v8f rr0 = z8, ii0 = z8, ri0 = z8, ir0 = z8;
    v8f rr1 = z8, ii1 = z8, ri1 = z8, ir1 = z8;
    const v2f* wt0 = wpk + (size_t)t0 * WFRAG;
    const v2f* wt1 = wpk + (size_t)t1 * WFRAG;
    for (int s = 0; s < KSTEPS; ++s) {
      int n0 = 4 * s + 2 * khalf;
      v2f are = *(const v2f*)&mre[m16 * NANT + n0];   // ds_load_b64 (shared)
      v2f aim = *(const v2f*)&mim[m16 * NANT + n0];
      v2f bre0 = wt0[(2 * s + 0) * 32 + lane];        // global b64, L2-hot
      v2f bim0 = wt0[(2 * s + 1) * 32 + lane];
      v2f bre1 = wt1[(2 * s + 0) * 32 + lane];
      v2f bim1 = wt1[(2 * s + 1) * 32 + lane];
      WMMA4(rr0, are, bre0);  WMMA4(ii0, aim, bim0);  // 8 independent chains
      WMMA4(ri0, are, bim0);  WMMA4(ir0, aim, bre0);
      WMMA4(rr1, are, bre1);  WMMA4(ii1, aim, bim1);
      WMMA4(ri1, are, bim1);  WMMA4(ir1, aim, bre1);
    }
    finalize_tile(md, t0, lane, rr0, ii0, ri0, ir0);
    finalize_tile(md, t1, lane, rr1, ii1, ri1, ir1);
  }
  if (wave == NWAVES - 1) {                  // remainder tile 74
    const int t = NTILES - 1;
    v8f rr = z8, ii = z8, ri = z8, ir = z8;
    const v2f* wt = wpk + (size_t)t * WFRAG;
    for (int s = 0; s < KSTEPS; ++s) {
      int n0 = 4 * s + 2 * khalf;
      v2f are = *(const v2f*)&mre[m16 * NANT + n0];
      v2f aim = *(const v2f*)&mim[m16 * NANT + n0];
      v2f bre = wt[(2 * s + 0) * 32 + lane];
      v2f bim = wt[(2 * s + 1) * 32 + lane];
      WMMA4(rr, are, bre);  WMMA4(ii, aim, bim);
      WMMA4(ri, are, bim);  WMMA4(ir, aim, bre);
    }
    finalize_tile(md, t, lane, rr, ii, ri, ir);
  }
  __syncthreads();

  // ---- Phase 2: matching pursuit, residual in registers (all threads) -----
  float rre[MDIM], rim[MDIM];
#pragma unroll
  for (int m = 0; m < MDIM; ++m) {
    float2 v = xg[(size_t)b * MDIM + m];
    rre[m] = v.x;
    rim[m] = v.y;
  }
  int kk = kp[0];
  if (kk < 0)  kk = 0;
  if (kk > 64) kk = 64;

  for (int it = 0; it < kk; ++it) {
    float bm = -1.f, bvr = 0.f, bvi = 0.f;
    int   bi = 0x7fffffff;
    for (int a = tid; a < AATOMS; a += THREADS) {
      const float4* col = (const float4*)&md[a * ASTRIDE];  // b128 loads
      float cr = 0.f, ci = 0.f;
#pragma unroll
      for (int j = 0; j < 8; ++j) {
        float4 w = col[j];          // {re(2j), im(2j), re(2j+1), im(2j+1)}
        int m0 = 2 * j, m1 = 2 * j + 1;
        cr += rre[m0] * w.x + rim[m0] * w.y + rre[m1] * w.z + rim[m1] * w.w;
        ci += rim[m0] * w.x - rre[m0] * w.y + rim[m1] * w.z - rre[m1] * w.w;
      }
      float mg = cr * cr + ci * ci;
      if (mg > bm) { bm = mg; bi = a; bvr = cr; bvi = ci; }
    }
    // wave32 argmax reduce (tie -> lower index, matches jnp.argmax)
#pragma unroll
    for (int off = 16; off >= 1; off >>= 1) {
      float om  = __shfl_xor(bm, off, 32);
      int   oi  = __shfl_xor(bi, off, 32);
      float ovr = __shfl_xor(bvr, off, 32);
      float ovi = __shfl_xor(bvi, off, 32);
      if (om > bm || (om == bm && oi < bi)) { bm = om; bi = oi; bvr = ovr; bvi = ovi; }
    }
    if (lane == 0) {
      red[wave * 4 + 0] = bm;
      red[wave * 4 + 1] = __int_as_float(bi);
      red[wave * 4 + 2] = bvr;
      red[wave * 4 + 3] = bvi;
    }
    __syncthreads();
    if (tid == 0) {
      float wm = red[0];
      int   wi = __float_as_int(red[1]);
      float wr = red[2], wv = red[3];
      for (int w2 = 1; w2 < NWAVES; ++w2) {
        float om = red[w2 * 4 + 0];
        int   oi = __float_as_int(red[w2 * 4 + 1]);
        if (om > wm || (om == wm && oi < wi)) {
          wm = om; wi = oi; wr = red[w2 * 4 + 2]; wv = red[w2 * 4 + 3];
        }
      }
      red[32] = __int_as_float(wi);
      red[33] = wr;
      red[34] = wv;
    }
    __syncthreads();
    int   wi = __float_as_int(red[32]);
    float vr = red[33], vi = red[34];
    const float4* col = (const float4*)&md[wi * ASTRIDE];  // broadcast read
#pragma unroll
    for (int j = 0; j < 8; ++j) {
      float4 w = col[j];
      int m0 = 2 * j, m1 = 2 * j + 1;
      rre[m0] -= vr * w.x - vi * w.y;        // residual -= val * atom
      rim[m0] -= vr * w.y + vi * w.x;
      rre[m1] -= vr * w.z - vi * w.w;
      rim[m1] -= vr * w.w + vi * w.z;
    }
    __syncthreads();                         // protect red[] reuse
  }

  // ---- outputs: residual then x_hat = x - residual, complex64 -------------
  if (tid < MDIM) {
    int m = tid;
    float2 r;
    r.x = rre[m];
    r.y = rim[m];
    outp[(size_t)b * MDIM + m] = r;
    float2 xv = xg[(size_t)b * MDIM + m];
    float2 h;
    h.x = xv.x - rre[m];
    h.y = xv.y - rim[m];
    outp[(size_t)N_B * MDIM + (size_t)b * MDIM + m] = h;
  }
}

// --------------------------------------------------------------------------
extern "C" void kernel_launch(void* const* d_in, const int* in_sizes, int n_in,
                              void* d_out, int out_size, void* d_ws, size_t ws_size,
                              hipStream_t stream) {
  (void)in_sizes; (void)n_in; (void)out_size; (void)ws_size;
  const float*  pos = (const float*)d_in[0];   // ant_position [64]
  const float*  doa = (const float*)d_in[1];   // DoA [1200]
  const float*  g   = (const float*)d_in[2];   // g_vec [64]
  const float*  lam = (const float*)d_in[3];   // lambda_ [1]
  const float2* x   = (const float2*)d_in[4];  // x [B,16] complex64
  const float2* M   = (const float2*)d_in[6];  // M [B,64,16] complex64
  const int*    kp  = (const int*)d_in[8];     // k (scalar)

  float2* wpk = (float2*)d_ws;                 // 75*16*2*32 float2 = 614400 B

  int nslots = NTILES * KSTEPS * 2 * 32;
  pack_w_kernel<<<(nslots + 255) / 256, 256, 0, stream>>>(pos, doa, g, lam, wpk);

  size_t smem = (size_t)SM_TOT * sizeof(float);  // 181248 B < 320 KB WGP LDS
  (void)hipFuncSetAttribute((const void*)mp_fused_kernel,
                            hipFuncAttributeMaxDynamicSharedMemorySize, (int)smem);
  mp_fused_kernel<<<N_B, THREADS, smem, stream>>>(M, x, (const v2f*)wpk, kp,
                                                  (float2*)d_out);
}